// EQ_81724637708929
// MI455X (gfx1250) — compile-verified
//
#include <hip/hip_runtime.h>
#include <math.h>
#include <stdint.h>

// Problem geometry: x is [8, 2, 1200000] f32, flattened = 16 channels x 1.2M.
#define NCHAN   16
#define TLEN    1200000
#define CHUNK   160                   // samples per chunk (160*4B = 640B, float4-aligned)
#define NCHUNK  7500                  // chunks per channel (1200000/160)
#define TOTCH   (NCHAN * NCHUNK)      // 120000 chunks total (channels are contiguous)
#define REGC    75                    // chunks per region
#define NREG    100                   // regions per channel (7500/75)
#define TOTREG  (NCHAN * NREG)        // 1600
#define EBLOCK  160                   // threads per block in apply kernel (5 waves), 120000/160=750 exact
#define LDS_STRIDE 164                // CHUNK + 4 pad dwords (656B, 16B aligned; 2-way bank conflict max)
#define EUNITS  (CHUNK / 4)           // 40 x b128 units per thread
#define QUARTER (CHUNK / 4)           // 40: 16B units per chunk

#if defined(__HIP_DEVICE_COMPILE__) && \
    __has_builtin(__builtin_amdgcn_global_load_async_to_lds_b128) && \
    __has_builtin(__builtin_amdgcn_global_store_async_from_lds_b128) && \
    __has_builtin(__builtin_amdgcn_s_wait_asynccnt)
#define USE_ASYNC 1
#else
#define USE_ASYNC 0
#endif

// The builtins take pointers to 16-byte int vectors (per hipcc diagnostic).
typedef int v4i_t __attribute__((vector_size(16)));
#define AS1CAST(p) ((__attribute__((address_space(1))) v4i_t*)(p))
#define AS3CAST(p) ((__attribute__((address_space(3))) v4i_t*)(p))

__device__ __forceinline__ float biq_step(float xv, float& x1, float& x2,
                                          float& y1, float& y2,
                                          float b0, float b1, float b2,
                                          float a1, float a2) {
    float f = b0 * xv + b1 * x1 + b2 * x2;
    float y = f - a1 * y1 - a2 * y2;
    x2 = x1; x1 = xv;
    y2 = y1; y1 = y;
    return y;
}

// Hot kernel (defined first so it leads the disasm): async-DMA region into padded
// LDS, run the recurrence with correct initial states (float4 LDS traffic),
// overwrite x with y in LDS, async-DMA back out.
//
// Staging map: 16B unit u = i*EBLOCK + tid covers chunk u/40 at dword offset
// (u%40)*4. Since EBLOCK = 4*40, u/40 = 4i + tid/40 and u%40 = tid%40, so both
// the LDS and global addresses are linear in i (no per-iteration division).
__global__ __launch_bounds__(EBLOCK) void eq_apply(
    const float* __restrict__ x, const float2* __restrict__ cinit,
    float* __restrict__ out,
    float b0, float b1, float b2, float a1, float a2)
{
    extern __shared__ float lds[];                       // EBLOCK * LDS_STRIDE floats
    const int tid = threadIdx.x;
    const long long regionBase = (long long)blockIdx.x * (EBLOCK * CHUNK); // dwords

    const int tq = tid / QUARTER;                        // 0..3
    const int tr = tid % QUARTER;                        // 0..39
    float* ldsStage = lds + tq * LDS_STRIDE + tr * 4;    // steps 4*LDS_STRIDE dwords per i
    const float* gIn = x + regionBase + tid * 4;         // steps EBLOCK*4 dwords per i

    // ---- stage in: region (EBLOCK*CHUNK dwords) global -> padded LDS ----
#if USE_ASYNC
#pragma unroll
    for (int i = 0; i < EUNITS; ++i) {
        __builtin_amdgcn_global_load_async_to_lds_b128(
            AS1CAST(gIn + i * (EBLOCK * 4)),
            AS3CAST(ldsStage + i * (4 * LDS_STRIDE)), 0, 0);
    }
    __builtin_amdgcn_s_wait_asynccnt(0);
    __syncthreads();
#else
    for (int i = 0; i < EUNITS; ++i) {
        float4 v = *(const float4*)(gIn + i * (EBLOCK * 4));
        *(float4*)(ldsStage + i * (4 * LDS_STRIDE)) = v;
    }
    __syncthreads();
#endif

    // ---- compute: each thread runs its chunk from LDS (b128 LDS ops) ----
    const int gchunk = blockIdx.x * EBLOCK + tid;
    const int k = gchunk % NCHUNK;
    const long long gbase = (long long)gchunk * CHUNK;
    float x1 = 0.f, x2 = 0.f;
    if (k != 0) { x1 = x[gbase - 1]; x2 = x[gbase - 2]; }
    float2 s = cinit[gchunk];
    float y1 = s.x, y2 = s.y;
    float4* my4 = (float4*)(lds + tid * LDS_STRIDE);     // 656B offset -> 16B aligned
#pragma unroll 4
    for (int i = 0; i < CHUNK / 4; ++i) {
        float4 v = my4[i];
        float4 r;
        r.x = biq_step(v.x, x1, x2, y1, y2, b0, b1, b2, a1, a2);
        r.y = biq_step(v.y, x1, x2, y1, y2, b0, b1, b2, a1, a2);
        r.z = biq_step(v.z, x1, x2, y1, y2, b0, b1, b2, a1, a2);
        r.w = biq_step(v.w, x1, x2, y1, y2, b0, b1, b2, a1, a2);
        my4[i] = r;
    }
    __syncthreads();

    // ---- stage out: padded LDS -> out (coalesced) ----
    float* gOut = out + regionBase + tid * 4;
#if USE_ASYNC
#pragma unroll
    for (int i = 0; i < EUNITS; ++i) {
        __builtin_amdgcn_global_store_async_from_lds_b128(
            AS1CAST(gOut + i * (EBLOCK * 4)),
            AS3CAST(ldsStage + i * (4 * LDS_STRIDE)), 0, 0);
    }
    __builtin_amdgcn_s_wait_asynccnt(0);
#else
    for (int i = 0; i < EUNITS; ++i) {
        float4 v = *(const float4*)(ldsStage + i * (4 * LDS_STRIDE));
        *(float4*)(gOut + i * (EBLOCK * 4)) = v;
    }
#endif
}

// Kernel 1: zero-state end state of each chunk. FIR halo uses real x (zeros at channel start).
__global__ __launch_bounds__(256) void eq_chunk_end(
    const float* __restrict__ x, float2* __restrict__ endA,
    float b0, float b1, float b2, float a1, float a2)
{
    int g = blockIdx.x * blockDim.x + threadIdx.x;
    if (g >= TOTCH) return;
    int k = g % NCHUNK;                       // chunk index within channel
    long long base = (long long)g * CHUNK;    // channels contiguous -> flat indexing
    float x1 = 0.f, x2 = 0.f;
    if (k != 0) { x1 = x[base - 1]; x2 = x[base - 2]; }
    float y1 = 0.f, y2 = 0.f;
    const float4* xv4 = (const float4*)(x + base);
#pragma unroll 4
    for (int i = 0; i < CHUNK / 4; ++i) {
        if ((i & 3) == 0)
            __builtin_prefetch(xv4 + i + 16, 0, 0);   // global_prefetch_b8, 256B ahead
        float4 v = xv4[i];
        biq_step(v.x, x1, x2, y1, y2, b0, b1, b2, a1, a2);
        biq_step(v.y, x1, x2, y1, y2, b0, b1, b2, a1, a2);
        biq_step(v.z, x1, x2, y1, y2, b0, b1, b2, a1, a2);
        biq_step(v.w, x1, x2, y1, y2, b0, b1, b2, a1, a2);
    }
    endA[g] = make_float2(y1, y2);            // (y[L-1], y[L-2])
}

// Kernel 2: fold REGC chunk transforms into one region zero-state end state.
__global__ __launch_bounds__(256) void eq_region(
    const float2* __restrict__ endA, float2* __restrict__ regEnd,
    float c00, float c01, float c10, float c11)
{
    int r = blockIdx.x * blockDim.x + threadIdx.x;
    if (r >= TOTREG) return;
    float s1 = 0.f, s2 = 0.f;
    for (int k = 0; k < REGC; ++k) {
        float2 e = endA[r * REGC + k];
        float n1 = e.x + c00 * s1 + c01 * s2;
        float n2 = e.y + c10 * s1 + c11 * s2;
        s1 = n1; s2 = n2;
    }
    regEnd[r] = make_float2(s1, s2);
}

// Kernel 3: serial scan over regions per channel (state resets at channel start).
__global__ __launch_bounds__(32) void eq_chanscan(
    const float2* __restrict__ regEnd, float2* __restrict__ regInit,
    float r00, float r01, float r10, float r11)
{
    int ch = blockIdx.x * blockDim.x + threadIdx.x;
    if (ch >= NCHAN) return;
    float s1 = 0.f, s2 = 0.f;
    for (int rr = 0; rr < NREG; ++rr) {
        int r = ch * NREG + rr;
        regInit[r] = make_float2(s1, s2);
        float2 e = regEnd[r];
        float n1 = e.x + r00 * s1 + r01 * s2;
        float n2 = e.y + r10 * s1 + r11 * s2;
        s1 = n1; s2 = n2;
    }
}

// Kernel 4: expand region-init states to per-chunk initial states.
__global__ __launch_bounds__(256) void eq_chunkinit(
    const float2* __restrict__ endA, const float2* __restrict__ regInit,
    float2* __restrict__ cinit,
    float c00, float c01, float c10, float c11)
{
    int r = blockIdx.x * blockDim.x + threadIdx.x;
    if (r >= TOTREG) return;
    float2 s = regInit[r];
    float s1 = s.x, s2 = s.y;
    for (int k = 0; k < REGC; ++k) {
        int g = r * REGC + k;
        cinit[g] = make_float2(s1, s2);
        float2 e = endA[g];
        float n1 = e.x + c00 * s1 + c01 * s2;
        float n2 = e.y + c10 * s1 + c11 * s2;
        s1 = n1; s2 = n2;
    }
}

extern "C" void kernel_launch(void* const* d_in, const int* in_sizes, int n_in,
                              void* d_out, int out_size, void* d_ws, size_t ws_size,
                              hipStream_t stream)
{
    const float* x = (const float*)d_in[0];
    float* out = (float*)d_out;
    char* ws = (char*)d_ws;
    float2* endA    = (float2*)(ws);
    float2* cinit   = (float2*)(ws + sizeof(float2) * (size_t)TOTCH);
    float2* regEnd  = (float2*)(ws + sizeof(float2) * (size_t)TOTCH * 2);
    float2* regInit = (float2*)(ws + sizeof(float2) * ((size_t)TOTCH * 2 + TOTREG));

    // ---- peaking-EQ coefficients (match reference: normalize by a0, store f32) ----
    const double PI = 3.14159265358979323846;
    double w0 = 2.0 * PI * 1000.0 / 44100.0;
    double A  = exp(6.0 / 40.0 * log(10.0));
    double alpha = sin(w0) / (2.0 * 0.707);
    double a0d = 1.0 + alpha / A;
    float b0 = (float)((1.0 + alpha * A) / a0d);
    float b1 = (float)((-2.0 * cos(w0)) / a0d);
    float b2 = (float)((1.0 - alpha * A) / a0d);
    float a1 = (float)((-2.0 * cos(w0)) / a0d);
    float a2 = (float)((1.0 - alpha / A) / a0d);

    // ---- homogeneous responses p[n], q[n] (use float-rounded a1,a2 to match device math)
    // C = M^CHUNK (chunk transfer), R = M^(CHUNK*REGC) (region transfer); filter is
    // stable (pole radius ~0.93) so these decay hard -> corrections are tiny.
    double da1 = (double)a1, da2 = (double)a2;
    double p1 = 1.0, p2 = 0.0, q1 = 0.0, q2 = 1.0;   // p[-1]=1,p[-2]=0 ; q[-1]=0,q[-2]=1
    float C[4] = {0, 0, 0, 0}, R[4] = {0, 0, 0, 0};
    for (int n = 0; n < CHUNK * REGC; ++n) {
        double p = -da1 * p1 - da2 * p2; p2 = p1; p1 = p;
        double q = -da1 * q1 - da2 * q2; q2 = q1; q1 = q;
        if (n == CHUNK - 1) { C[0] = (float)p1; C[1] = (float)q1; C[2] = (float)p2; C[3] = (float)q2; }
    }
    R[0] = (float)p1; R[1] = (float)q1; R[2] = (float)p2; R[3] = (float)q2;

    eq_chunk_end<<<(TOTCH + 255) / 256, 256, 0, stream>>>(x, endA, b0, b1, b2, a1, a2);
    eq_region<<<(TOTREG + 255) / 256, 256, 0, stream>>>(endA, regEnd, C[0], C[1], C[2], C[3]);
    eq_chanscan<<<1, 32, 0, stream>>>(regEnd, regInit, R[0], R[1], R[2], R[3]);
    eq_chunkinit<<<(TOTREG + 255) / 256, 256, 0, stream>>>(endA, regInit, cinit, C[0], C[1], C[2], C[3]);
    eq_apply<<<TOTCH / EBLOCK, EBLOCK, EBLOCK * LDS_STRIDE * sizeof(float), stream>>>(
        x, cinit, out, b0, b1, b2, a1, a2);
}